// NMA_31696858644717
// MI455X (gfx1250) — compile-verified
//
#include <hip/hip_runtime.h>
#include <math.h>

// MI455X (gfx1250, wave32). Problem is HBM-bandwidth bound (~768 MB moved,
// ~33 us at 23.3 TB/s). GEMMs (~6.4 MFLOP) run on the exact-fp32
// V_WMMA_F32_16X16X4_F32 path with the Euler ODE fused into the epilogue.

typedef __attribute__((ext_vector_type(2))) float v2f;
typedef __attribute__((ext_vector_type(8))) float v8f;

#define ODE_STEPS 9                 // T_STEPS - 1
#define ODE_DT (3.0f / 9.0f)        // T_END / (T_STEPS - 1)

__device__ __forceinline__ float euler_ode(float g) {
    float y = 0.5f;
#pragma unroll
    for (int t = 0; t < ODE_STEPS; ++t) {
        float s = sinf(y + g);
        y = y + ODE_DT * (-y + s * s);
    }
    return y;
}

// ---------------------------------------------------------------------------
// Kernel 1: fused axis pooling. One block per (b,c); streams the 64^3 volume
// once via b128 loads and emits z_w, z_h, z_d (64 floats each) and z_c.
// Thread tid: dq = tid&15 owns d = 4*dq..4*dq+3 (float4, coalesced), and
// g = tid>>4 owns the 4-wide w strip w = 4g..4g+3. The 16 threads sharing a
// w value are one contiguous half-wave, so z_w reduces with __shfl_xor only.
// ---------------------------------------------------------------------------
__global__ void __launch_bounds__(256)
pool_kernel(const float* __restrict__ x,
            float* __restrict__ z_w, float* __restrict__ z_h,
            float* __restrict__ z_d, float* __restrict__ z_c) {
    const int bc  = blockIdx.x;
    const int tid = threadIdx.x;
    const int dq  = tid & 15;                 // d quad: d = 4*dq + q
    const int g   = tid >> 4;                 // w strip: w = 4*g + wi
    const float4* __restrict__ x4 =
        (const float4*)(x + (size_t)bc * (64u * 64u * 64u));

    __shared__ float s_h[64], s_w[64], s_d[64];
    if (tid < 64) { s_h[tid] = 0.f; s_d[tid] = 0.f; }
    __syncthreads();

    float acc_d[4] = {0.f, 0.f, 0.f, 0.f};
    float acc_w[4] = {0.f, 0.f, 0.f, 0.f};

    for (int h = 0; h < 64; ++h) {
        float rowsum = 0.f;
#pragma unroll
        for (int wi = 0; wi < 4; ++wi) {
            const int w = g * 4 + wi;
            const float4 v = x4[((size_t)h * 64 + w) * 16 + dq];
            acc_d[0] += v.x; acc_d[1] += v.y;
            acc_d[2] += v.z; acc_d[3] += v.w;
            const float s = (v.x + v.y) + (v.z + v.w);
            acc_w[wi] += s;
            rowsum    += s;
        }
        // full-wave reduce, one LDS atomic per wave per h (8 per h)
#pragma unroll
        for (int off = 16; off > 0; off >>= 1)
            rowsum += __shfl_xor(rowsum, off, 32);
        if ((tid & 31) == 0) atomicAdd(&s_h[h], rowsum);
    }

    // z_w: reduce across the 16-lane half-wave owning this w (no atomics)
#pragma unroll
    for (int wi = 0; wi < 4; ++wi) {
        float vw = acc_w[wi];
#pragma unroll
        for (int off = 8; off > 0; off >>= 1)
            vw += __shfl_xor(vw, off, 32);
        if (dq == 0) s_w[g * 4 + wi] = vw;    // unique writer per w
    }
    // z_d: 16 g-groups fold into each d (one-time atomics)
#pragma unroll
    for (int q = 0; q < 4; ++q)
        atomicAdd(&s_d[dq * 4 + q], acc_d[q]);
    __syncthreads();

    const float inv = 1.0f / 4096.0f;
    if (tid < 64) {
        z_h[bc * 64 + tid] = s_h[tid] * inv;
        z_w[bc * 64 + tid] = s_w[tid] * inv;
        z_d[bc * 64 + tid] = s_d[tid] * inv;
    }
    if (tid == 0) {
        float tot = 0.f;
        for (int i = 0; i < 64; ++i) tot += s_d[i];
        z_c[bc] = tot * (1.0f / 262144.0f);
    }
}

// ---------------------------------------------------------------------------
// Kernel 2: fp32 WMMA GEMM (gamma = Z @ W) + fused Euler ODE epilogue.
// One wave per 16x16 D-tile, K stepped by 4 via V_WMMA_F32_16X16X4_F32.
// K/N are template constants (immediate address math); the channel path
// (M=2) masks out-of-range A rows with a VALU multiply — no EXEC branching,
// EXEC is all-ones for every WMMA.
//   A 16x4 : lane L: v2f = A[L%16][k0+2*(L>=16) .. +1]   (one b64 load)
//   B 4x16 : mirrored K striping across lane halves
//   C/D    : VGPR j, lane L -> row j+8*(L>=16), col L%16
// ---------------------------------------------------------------------------
template <int K, int N, bool BOUNDS>
__global__ void __launch_bounds__(32)
gemm_ode_kernel(const float* __restrict__ Z, const float* __restrict__ W,
                float* __restrict__ Aout, int M) {
    const int mt   = blockIdx.x / (N >> 4);
    const int nt   = blockIdx.x % (N >> 4);
    const int lane = threadIdx.x & 31;
    const int half = lane >> 4;          // 0 or 1
    const int r    = lane & 15;

    const int   row_a = mt * 16 + r;
    const int   row_c = BOUNDS ? min(row_a, M - 1) : row_a;
    const float amask = (!BOUNDS || row_a < M) ? 1.0f : 0.0f;
    const int   col_b = nt * 16 + r;

    const v2f* __restrict__ Zr = (const v2f*)(Z + (size_t)row_c * K);

    v8f acc = {};
#pragma unroll
    for (int k0 = 0; k0 < K; k0 += 4) {
        const int ka = k0 + 2 * half;    // even -> one b64 load for A
        v2f a = Zr[ka >> 1];
        if (BOUNDS) a *= amask;          // VALU mask, no EXEC divergence
        v2f b;
        b.x = W[(size_t)ka * N + col_b];
        b.y = W[(size_t)(ka + 1) * N + col_b];
        acc = __builtin_amdgcn_wmma_f32_16x16x4_f32(false, a, false, b,
                                                    (short)0, acc,
                                                    false, false);
    }

#pragma unroll
    for (int j = 0; j < 8; ++j) {
        const int row = mt * 16 + j + 8 * half;
        if (!BOUNDS || row < M)
            Aout[(size_t)row * N + nt * 16 + r] = euler_ode(acc[j]);
    }
}

// ---------------------------------------------------------------------------
// Kernel 3: out = x * A. One block per (b,c,i) slice (4096 floats), float4
// vectorized; profiles pre-scaled by softmax(alpha) and a_c into LDS so the
// inner loop is pure FMA. Matches reference axis placement: a_w -> dim2,
// a_h -> dim3, a_d -> dim4.
// ---------------------------------------------------------------------------
__global__ void __launch_bounds__(256)
apply_kernel(const float* __restrict__ x,
             const float* __restrict__ a_w, const float* __restrict__ a_h,
             const float* __restrict__ a_d, const float* __restrict__ a_c,
             const float* __restrict__ alpha, float* __restrict__ out) {
    const int bc  = blockIdx.x >> 6;
    const int i   = blockIdx.x & 63;     // dim2 index
    const int tid = threadIdx.x;

    // softmax over 3 logits (cheap, recomputed per thread)
    const float l0 = alpha[0], l1 = alpha[1], l2 = alpha[2];
    const float m  = fmaxf(l0, fmaxf(l1, l2));
    const float e0 = expf(l0 - m), e1 = expf(l1 - m), e2 = expf(l2 - m);
    const float is = 1.0f / (e0 + e1 + e2);

    const float ac = a_c[bc];
    const float base_term = (e0 * is) * a_w[bc * 64 + i] * ac;

    __shared__ float s_h[64];            // a1 * a_h * ac
    __shared__ float s_d[64];            // a2 * a_d * ac
    if (tid < 64) {
        s_h[tid] = (e1 * is) * a_h[bc * 64 + tid] * ac;
        s_d[tid] = (e2 * is) * a_d[bc * 64 + tid] * ac;
    }
    __syncthreads();

    const size_t slice = ((size_t)bc * 64 + i) * 4096u;
    const float4* __restrict__ x4 = (const float4*)(x + slice);
    float4* __restrict__ o4 = (float4*)(out + slice);

#pragma unroll
    for (int it = 0; it < 4; ++it) {
        const int idx = it * 256 + tid;  // 1024 float4 per slice
        const int j = idx >> 4;          // dim3 index
        const int k = (idx & 15) << 2;   // dim4 base
        const float4 v = x4[idx];
        const float t = base_term + s_h[j];
        float4 rv;
        rv.x = v.x * (t + s_d[k + 0]);
        rv.y = v.y * (t + s_d[k + 1]);
        rv.z = v.z * (t + s_d[k + 2]);
        rv.w = v.w * (t + s_d[k + 3]);
        o4[idx] = rv;
    }
}

// ---------------------------------------------------------------------------
extern "C" void kernel_launch(void* const* d_in, const int* in_sizes, int n_in,
                              void* d_out, int out_size, void* d_ws, size_t ws_size,
                              hipStream_t stream) {
    const float* x     = (const float*)d_in[0];
    const float* W_w   = (const float*)d_in[1];
    const float* W_h   = (const float*)d_in[2];
    const float* W_d   = (const float*)d_in[3];
    const float* W_c   = (const float*)d_in[4];
    const float* alpha = (const float*)d_in[5];
    float* out = (float*)d_out;

    // workspace layout (fp32): fully overwritten each call
    float* ws  = (float*)d_ws;
    float* z_w = ws;               // 16384
    float* z_h = ws + 16384;       // 16384
    float* z_d = ws + 32768;       // 16384
    float* z_c = ws + 49152;       //   256
    float* a_w = ws + 49408;       // 16384
    float* a_h = ws + 65792;       // 16384
    float* a_d = ws + 82176;       // 16384
    float* a_c = ws + 98560;       //   256   (total ~395 KB)

    // 1) fused pooling pass: one 256 MB read (b128 loads)
    pool_kernel<<<256, 256, 0, stream>>>(x, z_w, z_h, z_d, z_c);

    // 2) WMMA fp32 GEMM + fused Euler ODE
    //    spatial: M=256, K=64, N=64 -> 64 tiles, no bounds masking
    gemm_ode_kernel<64, 64, false><<<64, 32, 0, stream>>>(z_w, W_w, a_w, 256);
    gemm_ode_kernel<64, 64, false><<<64, 32, 0, stream>>>(z_h, W_h, a_h, 256);
    gemm_ode_kernel<64, 64, false><<<64, 32, 0, stream>>>(z_d, W_d, a_d, 256);
    //    channel: M=2 (padded to 16), K=128, N=128 -> 8 tiles, masked A rows
    gemm_ode_kernel<128, 128, true><<<8, 32, 0, stream>>>(z_c, W_c, a_c, 2);

    // 3) broadcast multiply: 256 MB read + 256 MB write
    apply_kernel<<<256 * 64, 256, 0, stream>>>(x, a_w, a_h, a_d, a_c, alpha, out);
}